// GCN_47244640256453
// MI455X (gfx1250) — compile-verified
//
#include <hip/hip_runtime.h>

// GCN (2x GraphConv, norm='both') for MI455X / gfx1250.
//
// Phases (all on `stream`):
//   1. zero  deg_src|deg_dst|agg            (contiguous ws region)
//   2. degree: per-edge atomic +1.0 into deg_src/deg_dst
//   3. norm  : deg -> rsqrt(max(deg,1))     (in place, both buffers at once)
//   4. scatter L1: wave-per-edge (lane=feature) gather x[src]*norm_src -> atomicAdd agg[dst]
//   5. gemm  L1: h = relu((agg * norm_dst) @ W1 + b1)   -- V_WMMA_F32_16X16X4_F32
//   6. zero  agg
//   7. scatter L2 with h
//   8. gemm  L2: out = relu((agg * norm_dst) @ W2 + b2) -- V_WMMA_F32_16X16X4_F32
//
// Roofline: edge phases dominate (2 x 51.2M f32 atomics + 2 x 205MB gathers, all
// hitting the 192MB L2 since node features are 12.8MB); GEMMs are ~0.3 GFLOP total,
// so fp32 WMMA (exact reference precision) is the right choice for them.

typedef float v2f __attribute__((ext_vector_type(2)));
typedef float v8f __attribute__((ext_vector_type(8)));

#define GCN_F_IN  32
#define GCN_F_HID 32
#define GCN_F_OUT 16

__global__ void gcn_zero_f32(float* __restrict__ p, int n) {
    int i = blockIdx.x * blockDim.x + threadIdx.x;
    int stride = gridDim.x * blockDim.x;
    for (; i < n; i += stride) p[i] = 0.0f;
}

__global__ void gcn_degree(const int* __restrict__ src, const int* __restrict__ dst,
                           float* __restrict__ deg_src, float* __restrict__ deg_dst,
                           int e_cnt) {
    int e = blockIdx.x * blockDim.x + threadIdx.x;
    if (e < e_cnt) {
        atomicAdd(&deg_src[src[e]], 1.0f);
        atomicAdd(&deg_dst[dst[e]], 1.0f);
    }
}

__global__ void gcn_norm(float* __restrict__ deg, int n) {
    int i = blockIdx.x * blockDim.x + threadIdx.x;
    if (i < n) deg[i] = rsqrtf(fmaxf(deg[i], 1.0f));
}

// One wave32 per edge; lane == feature column (F==32).
// Coalesced 128B gather from x[src], contiguous 128B atomic-add burst into agg[dst].
__global__ void gcn_scatter(const float* __restrict__ x, const int* __restrict__ src,
                            const int* __restrict__ dst, const float* __restrict__ norm_src,
                            float* __restrict__ agg, int e_cnt) {
    unsigned t = blockIdx.x * blockDim.x + threadIdx.x;
    unsigned e = t >> 5;
    unsigned f = t & 31u;
    if (e < (unsigned)e_cnt) {
        int s = src[e];
        int d = dst[e];
        float v = x[(size_t)s * 32 + f] * norm_src[s];
        atomicAdd(&agg[(size_t)d * 32 + f], v);
    }
}

// Y[n,FOUT] = relu((X[n,32] * norm_dst[n]) @ W[32,FOUT] + bias[FOUT])
// One wave per 16 rows. K=32 via 8 chained V_WMMA_F32_16X16X4_F32; CTILES 16-col tiles.
// Fragment layouts per CDNA5 ISA 7.12.2 (32-bit A 16x4 / B 4x16 / C 16x16).
// Tail handling is branch-free on the load side: row index is CLAMPED (A row m only
// affects output row m, and out-of-range rows are never stored), so the 8 fragment
// loads issue unpredicated in one clause. Stores use a wave-uniform full-tile test
// so the common path has zero per-element predication; EXEC is all-ones at each WMMA.
template <int FOUT, int CTILES>
__launch_bounds__(256)
__global__ void gcn_gemm_bias_relu(const float* __restrict__ X,
                                   const float* __restrict__ norm_dst,
                                   const float* __restrict__ W,
                                   const float* __restrict__ bias,
                                   float* __restrict__ Y, int n_rows) {
    const int lane  = threadIdx.x & 31;
    const int gwave = (int)((blockIdx.x * blockDim.x + threadIdx.x) >> 5);
    const int nwave = (int)((gridDim.x * blockDim.x) >> 5);

    const int m    = lane & 15;        // A: row in tile; B/C: column in tile
    const int koff = (lane >> 4) * 2;  // K sub-offset within a K=4 block (0 or 2)

    // Preload B fragments (weights) -- wave-invariant, kept in VGPRs.
    v2f   bfrag[CTILES][8];
    float bval[CTILES];
#pragma unroll
    for (int ct = 0; ct < CTILES; ++ct) {
#pragma unroll
        for (int kb = 0; kb < 8; ++kb) {
            int k = kb * 4 + koff;
            bfrag[ct][kb].x = W[(size_t)k * FOUT + ct * 16 + m];
            bfrag[ct][kb].y = W[(size_t)(k + 1) * FOUT + ct * 16 + m];
        }
        bval[ct] = bias[ct * 16 + m];
    }

    // row0 is wave-uniform -> EXEC is all-ones around every WMMA.
    for (int row0 = gwave * 16; row0 < n_rows; row0 += nwave * 16) {
        int row = row0 + m;
        if (row > n_rows - 1) row = n_rows - 1;  // clamp: always a valid load address

        const float  nd = norm_dst[row];
        const float* xp = X + (size_t)row * 32 + koff;

        // Issue all 8 A-fragment loads (8B each, 8B-aligned) before any use.
        v2f a[8];
#pragma unroll
        for (int kb = 0; kb < 8; ++kb) {
            v2f v = *(const v2f*)(xp + kb * 4);
            a[kb].x = v.x * nd;
            a[kb].y = v.y * nd;
        }

        v8f acc[CTILES] = {};  // zero accumulators
#pragma unroll
        for (int kb = 0; kb < 8; ++kb) {
#pragma unroll
            for (int ct = 0; ct < CTILES; ++ct) {
                // (neg_a, A, neg_b, B, c_mod, C, reuse_a, reuse_b)
                acc[ct] = __builtin_amdgcn_wmma_f32_16x16x4_f32(
                    false, a[kb], false, bfrag[ct][kb], (short)0, acc[ct], false, false);
            }
        }

        // C/D layout: acc[i] -> row (row0 + i + (lane<16?0:8)), col (ct*16 + m).
        // Wave-uniform full-tile test: fast path is branch-free (one base address,
        // consecutive rows fold into store-instruction offsets).
        const int mbase = (lane < 16) ? 0 : 8;
        float* yp = Y + (size_t)(row0 + mbase) * FOUT + m;
        if (row0 + 16 <= n_rows) {
#pragma unroll
            for (int ct = 0; ct < CTILES; ++ct) {
#pragma unroll
                for (int i = 0; i < 8; ++i) {
                    yp[(size_t)i * FOUT + ct * 16] =
                        fmaxf(acc[ct][i] + bval[ct], 0.0f);
                }
            }
        } else {  // tail tile (at most one per wave)
#pragma unroll
            for (int ct = 0; ct < CTILES; ++ct) {
#pragma unroll
                for (int i = 0; i < 8; ++i) {
                    if (row0 + mbase + i < n_rows) {
                        yp[(size_t)i * FOUT + ct * 16] =
                            fmaxf(acc[ct][i] + bval[ct], 0.0f);
                    }
                }
            }
        }
    }
}

extern "C" void kernel_launch(void* const* d_in, const int* in_sizes, int n_in,
                              void* d_out, int out_size, void* d_ws, size_t ws_size,
                              hipStream_t stream) {
    const float* x   = (const float*)d_in[0];
    const int*   src = (const int*)d_in[1];
    const int*   dst = (const int*)d_in[2];
    const float* W1  = (const float*)d_in[3];
    const float* b1  = (const float*)d_in[4];
    const float* W2  = (const float*)d_in[5];
    const float* b2  = (const float*)d_in[6];
    float*       out = (float*)d_out;

    const int n = in_sizes[0] / GCN_F_IN;  // 100000 nodes
    const int e = in_sizes[1];             // 1600000 edges

    // Workspace layout (floats): deg_src[n] | deg_dst[n] | agg[n*32] | h[n*32]
    float* deg_src = (float*)d_ws;
    float* deg_dst = deg_src + n;
    float* agg     = deg_dst + n;
    float* h       = agg + (size_t)n * 32;

    const int B = 256;

    // 1) zero deg_src, deg_dst, agg (contiguous region of n*34 floats)
    {
        int cnt = n * 34;
        gcn_zero_f32<<<(cnt + B - 1) / B, B, 0, stream>>>(deg_src, cnt);
    }
    // 2) degrees
    gcn_degree<<<(e + B - 1) / B, B, 0, stream>>>(src, dst, deg_src, deg_dst, e);
    // 3) norms (both buffers are contiguous -> one pass over 2n)
    gcn_norm<<<(2 * n + B - 1) / B, B, 0, stream>>>(deg_src, 2 * n);

    const unsigned scat_threads = (unsigned)e * 32u;
    const unsigned scat_blocks  = (scat_threads + B - 1) / B;
    const int gemm_waves  = (n + 15) / 16;
    const int gemm_blocks = (gemm_waves + (B / 32) - 1) / (B / 32);

    // 4) layer 1 scatter: agg += x[src] * norm_src[src]
    gcn_scatter<<<scat_blocks, B, 0, stream>>>(x, src, dst, deg_src, agg, e);
    // 5) layer 1 GEMM: h = relu((agg * norm_dst) @ W1 + b1)   [FOUT=32 -> 2 col tiles]
    gcn_gemm_bias_relu<GCN_F_HID, 2><<<gemm_blocks, B, 0, stream>>>(
        agg, deg_dst, W1, b1, h, n);

    // 6) re-zero agg for layer 2
    {
        int cnt = n * 32;
        gcn_zero_f32<<<(cnt + B - 1) / B, B, 0, stream>>>(agg, cnt);
    }
    // 7) layer 2 scatter with h
    gcn_scatter<<<scat_blocks, B, 0, stream>>>(h, src, dst, deg_src, agg, e);
    // 8) layer 2 GEMM: out = relu((agg * norm_dst) @ W2 + b2) [FOUT=16 -> 1 col tile]
    gcn_gemm_bias_relu<GCN_F_OUT, 1><<<gemm_blocks, B, 0, stream>>>(
        agg, deg_dst, W2, b2, out, n);
}